// TMSA_15814069584467
// MI455X (gfx1250) — compile-verified
//
#include <hip/hip_runtime.h>
#include <math.h>

// ---- problem constants (from reference) ----
#define BB    4
#define Dd    8
#define Hh    64
#define Ww    64
#define Cc    120          // channels
#define NHEAD 6
#define HDIM  20           // Cc / NHEAD
#define Nn    128          // tokens per window (2*8*8)
#define NWIN  1024         // total windows (BB * 256)
#define TOK   131072       // total tokens (BB*Dd*Hh*Ww == NWIN*Nn)
#define HIDD  240          // MLP hidden (2*Cc), also 2C for proj input
#define QKVW  360          // 3*Cc
#define ATTN_SCALE 0.22360679774997896f   // 20^-0.5

typedef float v2f __attribute__((ext_vector_type(2)));
typedef float v8f __attribute__((ext_vector_type(8)));

__device__ __forceinline__ v8f wmma4(v2f a, v2f b, v8f c) {
    return __builtin_amdgcn_wmma_f32_16x16x4_f32(false, a, false, b,
                                                 (short)0, c, false, false);
}

// =====================================================================
// Kernel 1: LayerNorm1 + roll(-1,-4,-4) + window partition
//   xw[widx*128+n][c] = LN(x[src]) ; xm = xw + pos_bias[n % 64][c]
// one block (128 thr) per windowed token
// =====================================================================
__global__ void k_ln_window(const float* __restrict__ x,
                            const float* __restrict__ g1,
                            const float* __restrict__ b1,
                            const float* __restrict__ pos,
                            float* __restrict__ xw,
                            float* __restrict__ xm)
{
    __shared__ float rsum[128];
    __shared__ float rsq[128];
    int tok  = blockIdx.x;
    int widx = tok >> 7;
    int n    = tok & 127;
    int b    = widx >> 8;
    int wl   = widx & 255;
    int wd = wl >> 6, wh = (wl >> 3) & 7, ww = wl & 7;
    int nd = n >> 6,  nh = (n >> 3) & 7,  nw = n & 7;
    int d = wd * 2 + nd, h = wh * 8 + nh, w = ww * 8 + nw;
    // inverse of roll by (-1,-4,-4): source = (d+1, h+4, w+4) mod dims
    int ds = (d + 1) & 7;
    int hs = (h + 4) & 63;
    int wsrc = (w + 4) & 63;
    const float* src = x + ((size_t)((b * Dd + ds) * Hh + hs) * Ww + wsrc) * Cc;

    int c = threadIdx.x;
    float v = (c < Cc) ? src[c] : 0.f;
    rsum[c] = v;
    rsq[c]  = v * v;
    __syncthreads();
    for (int s = 64; s > 0; s >>= 1) {
        if (c < s) { rsum[c] += rsum[c + s]; rsq[c] += rsq[c + s]; }
        __syncthreads();
    }
    float mu  = rsum[0] * (1.f / Cc);
    float var = rsq[0]  * (1.f / Cc) - mu * mu;
    float inv = rsqrtf(var + 1e-5f);
    if (c < Cc) {
        float ln = (v - mu) * inv * g1[c] + b1[c];
        size_t o = (size_t)tok * Cc + c;
        xw[o] = ln;
        xm[o] = ln + pos[(n & 63) * Cc + c];
    }
}

// =====================================================================
// Plain LayerNorm (LN2), one block per token
// =====================================================================
__global__ void k_ln2(const float* __restrict__ x,
                      const float* __restrict__ g,
                      const float* __restrict__ b,
                      float* __restrict__ out)
{
    __shared__ float rsum[128];
    __shared__ float rsq[128];
    size_t tok = blockIdx.x;
    const float* src = x + tok * Cc;
    int c = threadIdx.x;
    float v = (c < Cc) ? src[c] : 0.f;
    rsum[c] = v;
    rsq[c]  = v * v;
    __syncthreads();
    for (int s = 64; s > 0; s >>= 1) {
        if (c < s) { rsum[c] += rsum[c + s]; rsq[c] += rsq[c + s]; }
        __syncthreads();
    }
    float mu  = rsum[0] * (1.f / Cc);
    float var = rsq[0]  * (1.f / Cc) - mu * mu;
    float inv = rsqrtf(var + 1e-5f);
    if (c < Cc) out[tok * Cc + c] = (v - mu) * inv * g[c] + b[c];
}

// =====================================================================
// Generic WMMA f32 GEMM:  out[T x Nc] = A[T x K] * W[Nc x K]^T + bias
// Each wave register-blocks a 16x64 output strip (4 tiles, 4 v8f accs):
// one v2f A fragment load feeds 4 independent WMMAs per k-step.
// Launch: grid=(T/16), block = 32 * ceil(Nc/64); wave w covers cols
// [w*64, w*64+64).
// Operand layout (wave32, V_WMMA_F32_16X16X4_F32):
//   A 16x4 : lane 0-15 = M rows; v0/v1 = K {2*hi, 2*hi+1}
//   B 4x16 : lane&15 = N col;    v0/v1 = K {2*hi, 2*hi+1}
//   C/D    : vgpr r -> M = r + 8*hi ; N = lane&15
// =====================================================================
template <bool ACC>
__global__ void k_gemm_wmma(const float* __restrict__ A, int lda,
                            const float* __restrict__ Wt,
                            const float* __restrict__ bias,
                            float* __restrict__ out, int ldo,
                            int Nc, int K)
{
    int lane = threadIdx.x & 31;
    int wave = threadIdx.x >> 5;
    int m  = lane & 15;
    int hi = lane >> 4;
    int jb0 = wave * 64;
    if (jb0 >= Nc) return;                 // wave-uniform
    int mb = blockIdx.x * 16;

    const float* arow = A + (size_t)(mb + m) * lda;

    int col[4]; bool cv[4]; const float* wrow[4];
#pragma unroll
    for (int j = 0; j < 4; ++j) {
        col[j]  = jb0 + j * 16 + m;
        cv[j]   = (col[j] < Nc);
        wrow[j] = Wt + (size_t)(cv[j] ? col[j] : 0) * K;   // safe fallback row
    }

    v8f acc[4] = {};
    for (int k0 = 0; k0 < K; k0 += 4) {
        int ka = k0 + 2 * hi;
        v2f a = *(const v2f*)(arow + ka);               // global_load_b64
#pragma unroll
        for (int j = 0; j < 4; ++j) {
            v2f b = *(const v2f*)(wrow[j] + ka);        // global_load_b64
            acc[j] = wmma4(a, b, acc[j]);
        }
    }
#pragma unroll
    for (int j = 0; j < 4; ++j) {
        if (!cv[j]) continue;                           // post-WMMA, stores only
        float bv = bias[col[j]];
#pragma unroll
        for (int r = 0; r < 8; ++r) {
            size_t o = (size_t)(mb + r + 8 * hi) * ldo + col[j];
            float val = acc[j][r] + bv;
            if (ACC) val += out[o];
            out[o] = val;
        }
    }
}

// =====================================================================
// Self-attention: one block (256 thr = 8 waves) per (window, head)
//   S = scale*Q K^T + rpb[rpi] + mask ; softmax rows ; O = P V
//   writes x_out columns [120 + head*20 .. +20)
// Each wave owns exactly ONE 16-row tile in both matmul phases, so no
// fragment is loop-invariant -> bounded register pressure, no spills.
// =====================================================================
__global__ void k_attn_self(const float* __restrict__ qkv,
                            const float* __restrict__ rpb,
                            const int*   __restrict__ rpi,
                            const float* __restrict__ mask,
                            float* __restrict__ xout)
{
    __shared__ float sS[Nn * Nn];          // 64 KB
    int widx = blockIdx.x;
    int head = blockIdx.y;
    int lane = threadIdx.x & 31;
    int wave = threadIdx.x >> 5;           // 0..7 == row-tile mt
    int m  = lane & 15;
    int hi = lane >> 4;

    const float* base = qkv + (size_t)widx * Nn * QKVW;
    const float* Q  = base + head * HDIM;
    const float* Kp = base + Cc       + head * HDIM;
    const float* V  = base + 2 * Cc   + head * HDIM;
    const float* mw = mask + (size_t)(widx & 255) * Nn * Nn;

    // ---- S = scale * Q K^T + bias + mask : wave 'mt' computes rows [16mt,16mt+16) ----
    {
        int mt = wave;
        const float* qr = Q + (size_t)(mt * 16 + m) * QKVW;
        v8f acc[8] = {};
#pragma unroll
        for (int ks = 0; ks < 5; ++ks) {   // K = 20 = 5 * 4
            int ka = ks * 4 + 2 * hi;
            v2f a = *(const v2f*)(qr + ka);
#pragma unroll
            for (int nt = 0; nt < 8; ++nt) {
                v2f b = *(const v2f*)(Kp + (size_t)(nt * 16 + m) * QKVW + ka);
                acc[nt] = wmma4(a, b, acc[nt]);
            }
        }
#pragma unroll
        for (int nt = 0; nt < 8; ++nt) {
#pragma unroll
            for (int r = 0; r < 8; ++r) {
                int row  = mt * 16 + r + 8 * hi;
                int colk = nt * 16 + m;
                int idx  = row * Nn + colk;
                sS[idx] = acc[nt][r] * ATTN_SCALE
                        + rpb[rpi[idx] * NHEAD + head] + mw[idx];
            }
        }
    }
    __syncthreads();

    // ---- softmax: one thread per row ----
    if (threadIdx.x < Nn) {
        float* sr = sS + threadIdx.x * Nn;
        float mx = -1e30f;
        for (int j = 0; j < Nn; ++j) mx = fmaxf(mx, sr[j]);
        float sum = 0.f;
        for (int j = 0; j < Nn; ++j) { float e = __expf(sr[j] - mx); sr[j] = e; sum += e; }
        float inv = 1.f / sum;
        for (int j = 0; j < Nn; ++j) sr[j] *= inv;
    }
    __syncthreads();

    // ---- O = P V : wave 'mt' computes 16 rows x 2 col-tiles (20 cols padded to 32) ----
    {
        int mt = wave;
        v8f acc[2] = {};
        const float* pr = sS + (size_t)(mt * 16 + m) * Nn;
        for (int kb = 0; kb < Nn; kb += 4) {
            int ka = kb + 2 * hi;
            v2f a = *(const v2f*)(pr + ka);            // ds_load_b64
#pragma unroll
            for (int nt = 0; nt < 2; ++nt) {
                int colv = nt * 16 + m;
                bool cvld = (colv < HDIM);
                v2f b;
                b.x = cvld ? V[(size_t)ka * QKVW + colv]       : 0.f;
                b.y = cvld ? V[(size_t)(ka + 1) * QKVW + colv] : 0.f;
                acc[nt] = wmma4(a, b, acc[nt]);
            }
        }
#pragma unroll
        for (int nt = 0; nt < 2; ++nt) {
            int colv = nt * 16 + m;
            if (colv < HDIM) {
#pragma unroll
                for (int r = 0; r < 8; ++r) {
                    int orow = mt * 16 + r + 8 * hi;
                    xout[((size_t)widx * Nn + orow) * HIDD + Cc + head * HDIM + colv]
                        = acc[nt][r];
                }
            }
        }
    }
}

// =====================================================================
// Mutual attention: one block (128 thr = 4 waves) per (window, head, part)
//  part 0: queries = tokens 64..127, k/v = tokens 0..63  -> out rows 0..63
//  part 1: queries = tokens 0..63,  k/v = tokens 64..127 -> out rows 64..127
//  writes x_out columns [head*20 .. +20)
// One row-tile per wave in both phases (no hoistable fragments).
// =====================================================================
__global__ void k_attn_mut(const float* __restrict__ qkvm,
                           const float* __restrict__ mask,
                           float* __restrict__ xout)
{
    __shared__ float sS[64 * 64];          // 16 KB
    int widx = blockIdx.x;
    int head = blockIdx.y;
    int part = blockIdx.z;
    int lane = threadIdx.x & 31;
    int wave = threadIdx.x >> 5;           // 0..3 == row-tile mt
    int m  = lane & 15;
    int hi = lane >> 4;

    int qb = part ? 0 : 64;
    int kb = part ? 64 : 0;
    int ob = part ? 64 : 0;

    const float* base = qkvm + (size_t)widx * Nn * QKVW;
    const float* Q  = base + (size_t)qb * QKVW + head * HDIM;
    const float* Kp = base + (size_t)kb * QKVW + Cc     + head * HDIM;
    const float* V  = base + (size_t)kb * QKVW + 2 * Cc + head * HDIM;
    const float* mw = mask + (size_t)(widx & 255) * Nn * Nn;   // top-left 64x64

    // ---- S = scale * Q K^T + mask : wave 'mt' computes rows [16mt,16mt+16) ----
    {
        int mt = wave;
        const float* qr = Q + (size_t)(mt * 16 + m) * QKVW;
        v8f acc[4] = {};
#pragma unroll
        for (int ks = 0; ks < 5; ++ks) {
            int ka = ks * 4 + 2 * hi;
            v2f a = *(const v2f*)(qr + ka);
#pragma unroll
            for (int nt = 0; nt < 4; ++nt) {
                v2f b = *(const v2f*)(Kp + (size_t)(nt * 16 + m) * QKVW + ka);
                acc[nt] = wmma4(a, b, acc[nt]);
            }
        }
#pragma unroll
        for (int nt = 0; nt < 4; ++nt) {
#pragma unroll
            for (int r = 0; r < 8; ++r) {
                int row  = mt * 16 + r + 8 * hi;
                int colk = nt * 16 + m;
                sS[row * 64 + colk] = acc[nt][r] * ATTN_SCALE + mw[row * Nn + colk];
            }
        }
    }
    __syncthreads();

    // ---- softmax: one thread per row (64 rows) ----
    if (threadIdx.x < 64) {
        float* sr = sS + threadIdx.x * 64;
        float mx = -1e30f;
        for (int j = 0; j < 64; ++j) mx = fmaxf(mx, sr[j]);
        float sum = 0.f;
        for (int j = 0; j < 64; ++j) { float e = __expf(sr[j] - mx); sr[j] = e; sum += e; }
        float inv = 1.f / sum;
        for (int j = 0; j < 64; ++j) sr[j] *= inv;
    }
    __syncthreads();

    // ---- O = P V : wave 'mt' computes 16 rows x 2 col-tiles ----
    {
        int mt = wave;
        v8f acc[2] = {};
        const float* pr = sS + (size_t)(mt * 16 + m) * 64;
        for (int kk = 0; kk < 64; kk += 4) {
            int ka = kk + 2 * hi;
            v2f a = *(const v2f*)(pr + ka);            // ds_load_b64
#pragma unroll
            for (int nt = 0; nt < 2; ++nt) {
                int colv = nt * 16 + m;
                bool cvld = (colv < HDIM);
                v2f b;
                b.x = cvld ? V[(size_t)ka * QKVW + colv]       : 0.f;
                b.y = cvld ? V[(size_t)(ka + 1) * QKVW + colv] : 0.f;
                acc[nt] = wmma4(a, b, acc[nt]);
            }
        }
#pragma unroll
        for (int nt = 0; nt < 2; ++nt) {
            int colv = nt * 16 + m;
            if (colv < HDIM) {
#pragma unroll
                for (int r = 0; r < 8; ++r) {
                    int orow = mt * 16 + r + 8 * hi;
                    xout[((size_t)widx * Nn + ob + orow) * HIDD + head * HDIM + colv]
                        = acc[nt][r];
                }
            }
        }
    }
}

// =====================================================================
// Window reverse + roll(+1,+4,+4) + residual:  out = x + y
// =====================================================================
__global__ void k_unwindow_add(const float* __restrict__ x,
                               const float* __restrict__ ywin,
                               float* __restrict__ out)
{
    size_t i = (size_t)blockIdx.x * blockDim.x + threadIdx.x;
    size_t total = (size_t)TOK * Cc;
    if (i >= total) return;
    int c = (int)(i % Cc);
    size_t t = i / Cc;
    int w = (int)(t % Ww); t /= Ww;
    int h = (int)(t % Hh); t /= Hh;
    int d = (int)(t % Dd);
    int b = (int)(t / Dd);
    int dr = (d + 7) & 7;      // d-1 mod 8
    int hr = (h + 60) & 63;    // h-4 mod 64
    int wr = (w + 60) & 63;    // w-4 mod 64
    int widx = ((b * 4 + (dr >> 1)) * 8 + (hr >> 3)) * 8 + (wr >> 3);
    int n = ((dr & 1) << 6) | ((hr & 7) << 3) | (wr & 7);
    out[i] = x[i] + ywin[((size_t)widx * Nn + n) * Cc + c];
}

// =====================================================================
// GEGLU combine: o = gelu_exact(a) * b
// =====================================================================
__global__ void k_gelu_mul(const float* __restrict__ a,
                           const float* __restrict__ b,
                           float* __restrict__ o, size_t n)
{
    size_t i = (size_t)blockIdx.x * blockDim.x + threadIdx.x;
    if (i >= n) return;
    float v = a[i];
    float g = 0.5f * v * (1.f + erff(v * 0.70710678118654752f));
    o[i] = g * b[i];
}

// =====================================================================
extern "C" void kernel_launch(void* const* d_in, const int* in_sizes, int n_in,
                              void* d_out, int out_size, void* d_ws, size_t ws_size,
                              hipStream_t stream)
{
    const float* x     = (const float*)d_in[0];
    const float* amask = (const float*)d_in[1];
    const float* g1    = (const float*)d_in[2];
    const float* b1    = (const float*)d_in[3];
    const float* qsw   = (const float*)d_in[4];
    const float* qsb   = (const float*)d_in[5];
    const float* qmw   = (const float*)d_in[6];
    const float* qmb   = (const float*)d_in[7];
    const float* pw    = (const float*)d_in[8];
    const float* pb    = (const float*)d_in[9];
    const float* rpb   = (const float*)d_in[10];
    const float* pos   = (const float*)d_in[11];
    const float* g2    = (const float*)d_in[12];
    const float* b2    = (const float*)d_in[13];
    const float* f11w  = (const float*)d_in[14];
    const float* f11b  = (const float*)d_in[15];
    const float* f12w  = (const float*)d_in[16];
    const float* f12b  = (const float*)d_in[17];
    const float* f2w   = (const float*)d_in[18];
    const float* f2b   = (const float*)d_in[19];
    const int*   rpi   = (const int*)d_in[20];
    float* outf = (float*)d_out;

    // workspace layout (floats), with dead-buffer reuse
    float* ws = (float*)d_ws;
    float* xw = ws;                               // TOK*120
    float* xm = xw + (size_t)TOK * Cc;            // TOK*120
    float* qs = xm + (size_t)TOK * Cc;            // TOK*360
    float* qm = qs + (size_t)TOK * QKVW;          // TOK*360
    float* xo = qm + (size_t)TOK * QKVW;          // TOK*240
    float* ywin = xw;   // reuse (xw dead after qkv_self gemm)
    float* z    = xm;   // reuse (xm dead after qkv_mut gemm)
    float* h1   = qs;   // reuse (qs dead after self-attn)
    float* h2   = qm;   // reuse (qm dead after mutual-attn)
    float* hp   = xo;   // reuse (xo dead after proj gemm)

    // 1) LN1 + roll + window partition (+ pos-bias variant)
    k_ln_window<<<TOK, 128, 0, stream>>>(x, g1, b1, pos, xw, xm);

    // 2) qkv projections (WMMA f32); block = 32 * ceil(Nc/64) threads
    k_gemm_wmma<false><<<TOK / 16, 192, 0, stream>>>(xw, Cc, qsw, qsb, qs, QKVW, QKVW, Cc);
    k_gemm_wmma<false><<<TOK / 16, 192, 0, stream>>>(xm, Cc, qmw, qmb, qm, QKVW, QKVW, Cc);

    // 3) attention (WMMA f32) -> concat buffer xo[TOK x 240]
    k_attn_self<<<dim3(NWIN, NHEAD), 256, 0, stream>>>(qs, rpb, rpi, amask, xo);
    k_attn_mut<<<dim3(NWIN, NHEAD, 2), 128, 0, stream>>>(qm, amask, xo);

    // 4) output projection (Nc=120 -> 2 waves)
    k_gemm_wmma<false><<<TOK / 16, 64, 0, stream>>>(xo, HIDD, pw, pb, ywin, Cc, Cc, HIDD);

    // 5) window reverse + roll back + residual -> d_out (intermediate x)
    {
        size_t total = (size_t)TOK * Cc;
        k_unwindow_add<<<(unsigned)((total + 255) / 256), 256, 0, stream>>>(x, ywin, outf);
    }

    // 6) LN2 + GEGLU MLP, fc2 accumulates residual into d_out
    k_ln2<<<TOK, 128, 0, stream>>>(outf, g2, b2, z);
    k_gemm_wmma<false><<<TOK / 16, 128, 0, stream>>>(z, Cc, f11w, f11b, h1, HIDD, HIDD, Cc);
    k_gemm_wmma<false><<<TOK / 16, 128, 0, stream>>>(z, Cc, f12w, f12b, h2, HIDD, HIDD, Cc);
    {
        size_t total = (size_t)TOK * HIDD;
        k_gelu_mul<<<(unsigned)((total + 255) / 256), 256, 0, stream>>>(h1, h2, hp, total);
    }
    k_gemm_wmma<true><<<TOK / 16, 64, 0, stream>>>(hp, HIDD, f2w, f2b, outf, Cc, Cc, HIDD);
}